// GenericLinear_89300960018587
// MI455X (gfx1250) — compile-verified
//
#include <hip/hip_runtime.h>

typedef __attribute__((ext_vector_type(2))) float v2f;
typedef __attribute__((ext_vector_type(4))) float v4f;
typedef __attribute__((ext_vector_type(8))) float v8f;

#if __has_builtin(__builtin_amdgcn_global_load_async_to_lds_b128) && \
    __has_builtin(__builtin_amdgcn_s_wait_asynccnt)
#define USE_ASYNC 1
#else
#define USE_ASYNC 0
#endif

// Builtin parameter types per hipcc diagnostic: int __vector_size__(16),
// addrspace(1) for global (prints as __device__*), addrspace(3) for LDS.
typedef int v4i_ __attribute__((vector_size(4 * sizeof(int))));
typedef __attribute__((address_space(1))) v4i_ gv4i_t;   // global int4
typedef __attribute__((address_space(3))) v4i_ lv4i_t;   // LDS int4

#define NBLADES 4
#define BSZ     16384            // batch rows per blade
#define CIN     256
#define UOUT    256
#define WROW    (NBLADES * UOUT) // 1024: row stride of W

#define BM 128
#define BN 128
#define BK 32
#define LDA 36    // BK + 4 pad floats: 16B-aligned rows, conflict-free b64 frag reads
#define LDB 144   // BN + 16 pad floats: 16B-aligned rows, lane-half banks disjoint
#define NCHUNK (NBLADES * (CIN / BK))   // 32 K-chunks total (K_eff = 1024)
#define ASYNC_PER_CHUNK 8               // 4 (A) + 4 (B) b128 asyncs per thread

// j(i,k): weight blade feeding output blade k from input blade i; s(i,k): sign
__device__ __constant__ int   c_jmap[4][4] = {{0,1,2,3},{1,0,3,2},{2,3,0,1},{3,2,1,0}};
__device__ __constant__ float c_smap[4][4] = {{ 1.f, 1.f, 1.f, 1.f},
                                              {-1.f, 1.f,-1.f, 1.f},
                                              {-1.f, 1.f, 1.f,-1.f},
                                              {-1.f,-1.f, 1.f, 1.f}};

// Stage one BK-chunk of A (128x32) and B (32x128) into LDS: pure byte copy,
// sign is handled by accumulator negation at blade-phase boundaries.
__device__ __forceinline__ void stage_chunk(int c, int kOut, int tid,
                                            const float* __restrict__ x,
                                            const float* __restrict__ W,
                                            int rowBase, int colBase,
                                            float* sA, float* sB)
{
    const int ib = c >> 3;
    const int kc = (c & 7) * BK;
    const int jb = c_jmap[ib][kOut];
    const float* __restrict__ xblk = x + (size_t)(ib * BSZ + rowBase) * CIN + kc;
    const float* __restrict__ wblk = W + (size_t)kc * WROW + jb * UOUT + colBase;

    #pragma unroll
    for (int t = 0; t < 4; ++t) {            // A: 1024 float4 slots
        const int lin = tid + t * 256;
        const int ar  = lin >> 3;            // row 0..127
        const int ac  = (lin & 7) << 2;      // col 0..28 step 4
        const float* gp = xblk + (size_t)ar * CIN + ac;
        float*       lp = sA + ar * LDA + ac;
#if USE_ASYNC
        __builtin_amdgcn_global_load_async_to_lds_b128((gv4i_t*)gp, (lv4i_t*)lp, 0, 0);
#else
        *(v4f*)lp = *(const v4f*)gp;
#endif
    }
    #pragma unroll
    for (int t = 0; t < 4; ++t) {            // B: 1024 float4 slots
        const int lin = tid + t * 256;
        const int br  = lin >> 5;            // K-row 0..31
        const int bc  = (lin & 31) << 2;     // col 0..124 step 4
        const float* gp = wblk + (size_t)br * WROW + bc;
        float*       lp = sB + br * LDB + bc;
#if USE_ASYNC
        __builtin_amdgcn_global_load_async_to_lds_b128((gv4i_t*)gp, (lv4i_t*)lp, 0, 0);
#else
        *(v4f*)lp = *(const v4f*)gp;
#endif
    }
}

__global__ __launch_bounds__(256)
void ga_dense_wmma_f32(const float* __restrict__ x,
                       const float* __restrict__ W,
                       const float* __restrict__ b,
                       float* __restrict__ out)
{
    __shared__ float shA[2][BM * LDA];   // double-buffered: 2 x 18432 B
    __shared__ float shB[2][BK * LDB];   // double-buffered: 2 x 18432 B

    const int tid  = threadIdx.x;
    const int lane = tid & 31;
    const int wid  = tid >> 5;        // 0..7 (8 wave32 waves)
    const int half = lane >> 4;       // lane-half: 0 or 1
    const int lr   = lane & 15;

    const int rowBase = blockIdx.x * BM;   // within one blade's batch block
    const int colBase = blockIdx.y * BN;   // within U
    const int kOut    = blockIdx.z;        // output blade

    const int wm = wid >> 1;          // 4x2 wave grid; wave subtile 32x64
    const int wn = wid & 1;
    const int m0 = wm * 32;
    const int n0 = wn * 64;

    v8f acc[2][4];
    #pragma unroll
    for (int mt = 0; mt < 2; ++mt)
        #pragma unroll
        for (int nt = 0; nt < 4; ++nt)
            acc[mt][nt] = (v8f){};

    stage_chunk(0, kOut, tid, x, W, rowBase, colBase, &shA[0][0], &shB[0][0]);

    float sprev = 1.f;   // acc holds sprev * (true signed partial sum)

    for (int c = 0; c < NCHUNK; ++c) {
        const int buf = c & 1;
        if (c + 1 < NCHUNK)
            stage_chunk(c + 1, kOut, tid, x, W, rowBase, colBase,
                        &shA[buf ^ 1][0], &shB[buf ^ 1][0]);
#if USE_ASYNC
        if (c + 1 < NCHUNK) __builtin_amdgcn_s_wait_asynccnt(ASYNC_PER_CHUNK);
        else                __builtin_amdgcn_s_wait_asynccnt(0);
#endif
        __syncthreads();   // chunk c resident in LDS for all waves

        // blade-phase boundary: fold Cayley sign by negating the accumulators
        if ((c & 7) == 0) {
            const float s = c_smap[c >> 3][kOut];
            if (s * sprev < 0.f) {
                #pragma unroll
                for (int mt = 0; mt < 2; ++mt)
                    #pragma unroll
                    for (int nt = 0; nt < 4; ++nt)
                        acc[mt][nt] = -acc[mt][nt];
            }
            sprev = s;
        }

        const float* sA = &shA[buf][0];
        const float* sB = &shB[buf][0];
        #pragma unroll
        for (int kk = 0; kk < BK; kk += 4) {      // 8 WMMA K-steps of 4
            const int kA = kk + 2 * half;         // ISA layout: VGPR t, half h -> K = 2h + t
            v2f afrag[2];
            #pragma unroll
            for (int mt = 0; mt < 2; ++mt)
                afrag[mt] = *(const v2f*)(&sA[(m0 + mt * 16 + lr) * LDA + kA]);
            #pragma unroll
            for (int nt = 0; nt < 4; ++nt) {
                const int col = n0 + nt * 16 + lr;
                v2f bfrag;
                bfrag.x = sB[(kA + 0) * LDB + col];
                bfrag.y = sB[(kA + 1) * LDB + col];
                #pragma unroll
                for (int mt = 0; mt < 2; ++mt)
                    acc[mt][nt] = __builtin_amdgcn_wmma_f32_16x16x4_f32(
                        false, afrag[mt], false, bfrag,
                        (short)0, acc[mt][nt], false, false);
            }
        }
        __syncthreads();   // protect buf before it is re-staged next iteration
    }

    // epilogue: out = s(3,k) * acc + bias_k   (C layout: vgpr v, half h -> M = v + 8h)
    const float* __restrict__ bk = b + kOut * UOUT + colBase;
    float* __restrict__ outblk   = out + (size_t)(kOut * BSZ + rowBase) * UOUT + colBase;
    #pragma unroll
    for (int nt = 0; nt < 4; ++nt) {
        const int col   = n0 + nt * 16 + lr;
        const float bia = bk[col];
        #pragma unroll
        for (int mt = 0; mt < 2; ++mt) {
            #pragma unroll
            for (int v = 0; v < 8; ++v) {
                const int row = m0 + mt * 16 + half * 8 + v;
                outblk[(size_t)row * UOUT + col] = sprev * acc[mt][nt][v] + bia;
            }
        }
    }
}

extern "C" void kernel_launch(void* const* d_in, const int* in_sizes, int n_in,
                              void* d_out, int out_size, void* d_ws, size_t ws_size,
                              hipStream_t stream) {
    (void)in_sizes; (void)n_in; (void)out_size; (void)d_ws; (void)ws_size;
    const float* x = (const float*)d_in[0];
    const float* W = (const float*)d_in[1];
    const float* b = (const float*)d_in[2];
    float* out = (float*)d_out;
    dim3 grid(BSZ / BM, UOUT / BN, NBLADES);  // (128, 2, 4) = 1024 workgroups
    dim3 block(256, 1, 1);                    // 8 wave32 waves
    hipLaunchKernelGGL(ga_dense_wmma_f32, grid, block, 0, stream, x, W, b, out);
}